// QueryAndGroupLocalRelation_56822417326467
// MI455X (gfx1250) — compile-verified
//
#include <hip/hip_runtime.h>

// CDNA5 / gfx1250 ball-query + grouping, WMMA-accelerated distance tiles.
//
// d2(q,p) = |q|^2 - 2 q.p + |p|^2  < r^2
//   A (16x4 f32)  row  = ( qx, qy, qz, 1 )
//   B (4x16 f32)  col  = ( -2px, -2py, -2pz, |p|^2 )
//   D = A*B + 0   =>  D[m][n] = -2 q_m . p_n + |p_n|^2
//   compare D[m][n] < th_m where th_m = r^2 - |q_m|^2
// One V_WMMA_F32_16X16X4_F32 per 16x16 distance tile; B fragments are
// software-prefetched one tile ahead; empty tiles are skipped by a
// uniform (SGPR ballot) branch.

typedef __attribute__((ext_vector_type(2))) float v2f;
typedef __attribute__((ext_vector_type(8))) float v8f;

#define RADIUS2   4.0f
#define NS        32      // nsample
#define CCH       20      // channels
#define QPW       16      // queries per wave (== WMMA M)
#define WAVES     8
#define QPB       (QPW * WAVES)   // 128 queries per block
#define CHUNK     2048    // points staged in LDS per pass
#define TILE      16      // WMMA N

__device__ __forceinline__ float lane_bcast_f32(float v, int srcLane) {
    int r = __builtin_amdgcn_ds_bpermute(srcLane << 2, __float_as_int(v));
    return __int_as_float(r);
}

__global__ __launch_bounds__(256) void ballquery_group_wmma(
    const float* __restrict__ xyz,      // (nb*N, 3) points
    const float* __restrict__ new_xyz,  // (nb*N, 3) queries
    const float* __restrict__ pred,     // (nb*N, C)
    const float* __restrict__ tgt,      // (nb*N, C)
    float* __restrict__ out,            // [predLR | tgtLR | weight]
    int N, int nb, int blocksPerBatch)
{
    __shared__ float4 ptsLds[CHUNK];        // {-2x, -2y, -2z, |p|^2}
    __shared__ int    idxLds[QPB * NS];
    __shared__ int    cntLds[QPB];

    const int tid   = threadIdx.x;
    const int lane  = tid & 31;
    const int wave  = tid >> 5;
    const int batch = blockIdx.x / blocksPerBatch;
    const int blkIB = blockIdx.x % blocksPerBatch;
    const int qLocalBase = blkIB * QPB;
    const int base  = batch * N;            // global base of this batch segment

    // ---- per-wave query setup (row r = lane & 15 of the 16x16 D tile) ----
    const int row    = lane & 15;
    const int qLocal = qLocalBase + wave * QPW + row;
    const int qClamp = (qLocal < N) ? qLocal : (N - 1);
    const float qx = new_xyz[(size_t)(base + qClamp) * 3 + 0];
    const float qy = new_xyz[(size_t)(base + qClamp) * 3 + 1];
    const float qz = new_xyz[(size_t)(base + qClamp) * 3 + 2];
    const float thRow = RADIUS2 - (qx * qx + qy * qy + qz * qz);

    // A fragment: lanes 0-15 hold K0/K1 (qx,qy), lanes 16-31 hold K2/K3 (qz,1)
    v2f afrag;
    afrag.x = (lane < 16) ? qx : qz;
    afrag.y = (lane < 16) ? qy : 1.0f;

    // thresholds arranged in D layout: vgpr v, lane l -> row v + 8*(l>>4)
    const int hi8 = (lane >> 4) << 3;
    v8f th;
    th[0] = lane_bcast_f32(thRow, 0 + hi8);
    th[1] = lane_bcast_f32(thRow, 1 + hi8);
    th[2] = lane_bcast_f32(thRow, 2 + hi8);
    th[3] = lane_bcast_f32(thRow, 3 + hi8);
    th[4] = lane_bcast_f32(thRow, 4 + hi8);
    th[5] = lane_bcast_f32(thRow, 5 + hi8);
    th[6] = lane_bcast_f32(thRow, 6 + hi8);
    th[7] = lane_bcast_f32(thRow, 7 + hi8);

    int cnt = 0;  // neighbors found so far (valid on lanes 0-15)
    const int rb = (wave * QPW + row) * NS;   // this row's idxLds base

    // ---- scan the batch segment in LDS-staged chunks, ascending order ----
    for (int cb = 0; cb < N; cb += CHUNK) {
        __syncthreads();
        for (int j = tid; j < CHUNK; j += blockDim.x) {
            const int pj = cb + j;
            float4 v;
            if (pj < N) {
                const float px = xyz[(size_t)(base + pj) * 3 + 0];
                const float py = xyz[(size_t)(base + pj) * 3 + 1];
                const float pz = xyz[(size_t)(base + pj) * 3 + 2];
                v = make_float4(-2.0f * px, -2.0f * py, -2.0f * pz,
                                px * px + py * py + pz * pz);
            } else {
                v = make_float4(0.0f, 0.0f, 0.0f, 3.0e38f); // never selected
            }
            ptsLds[j] = v;
        }
        __syncthreads();

        const int tmax = (N - cb < CHUNK) ? (N - cb) : CHUNK;
        // per-lane B-fragment pointer: tile tt fragment lives at bp[tt*2]
        const float2* bp =
            (const float2*)ptsLds + 2 * row + (lane >> 4);

        // software pipeline: prefetch B fragment one tile ahead
        float2 bb = bp[0];
#pragma unroll 2
        for (int tt = 0; tt < tmax; tt += TILE) {
            const int ttn = tt + TILE;
            const float2 bbn = bp[((ttn < tmax) ? ttn : 0) * 2]; // prefetch

            v2f bfrag;
            bfrag.x = bb.x;
            bfrag.y = bb.y;

            v8f cz = {};
            v8f d = __builtin_amdgcn_wmma_f32_16x16x4_f32(
                false, afrag, false, bfrag, (short)0, cz, false, false);

            // D vgpr v: bits 0-15 = row v, bits 16-31 = row v+8 (cols 0..15)
            const unsigned b0 = __builtin_amdgcn_ballot_w32(d[0] < th[0]);
            const unsigned b1 = __builtin_amdgcn_ballot_w32(d[1] < th[1]);
            const unsigned b2 = __builtin_amdgcn_ballot_w32(d[2] < th[2]);
            const unsigned b3 = __builtin_amdgcn_ballot_w32(d[3] < th[3]);
            const unsigned b4 = __builtin_amdgcn_ballot_w32(d[4] < th[4]);
            const unsigned b5 = __builtin_amdgcn_ballot_w32(d[5] < th[5]);
            const unsigned b6 = __builtin_amdgcn_ballot_w32(d[6] < th[6]);
            const unsigned b7 = __builtin_amdgcn_ballot_w32(d[7] < th[7]);

            // uniform (scalar) skip of empty tiles: the common case
            const unsigned any = (b0 | b1) | (b2 | b3) | (b4 | b5) | (b6 | b7);
            if (any != 0u) {
                if (lane < 16) {
                    // select tree: pick ballot[row & 7], then half by row & 8
                    unsigned s01 = (row & 1) ? b1 : b0;
                    unsigned s23 = (row & 1) ? b3 : b2;
                    unsigned s45 = (row & 1) ? b5 : b4;
                    unsigned s67 = (row & 1) ? b7 : b6;
                    unsigned s03 = (row & 2) ? s23 : s01;
                    unsigned s47 = (row & 2) ? s67 : s45;
                    unsigned s07 = (row & 4) ? s47 : s03;
                    unsigned m16 = (row & 8) ? (s07 >> 16) : (s07 & 0xFFFFu);

                    const int gb = cb + tt;
                    while (m16 && cnt < NS) {
                        const int b = (int)__builtin_ctz(m16);
                        idxLds[rb + cnt] = gb + b;   // ascending index order
                        ++cnt;
                        m16 &= m16 - 1;
                    }
                }
            }
            bb = bbn;
        }
    }

    // ---- pad unfilled slots with first index; record count ----
    if (lane < 16) {
        if (cnt == 0) { idxLds[rb] = qClamp; cnt = 1; } // self always in range
        const int first = idxLds[rb];
        for (int s = cnt; s < NS; ++s) idxLds[rb + s] = first;
        cntLds[wave * QPW + row] = cnt;
    }
    __syncthreads();

    // ---- output phase: relations + weights ----
    const int m = nb * N;
    float* __restrict__ predOut = out;
    float* __restrict__ tgtOut  = out + (size_t)m * CCH * NS;
    float* __restrict__ wOut    = out + 2ull * (size_t)m * CCH * NS;

    for (int pair = tid; pair < QPB * NS; pair += blockDim.x) {
        const int q  = pair >> 5;   // NS == 32
        const int s  = pair & 31;
        const int ql = qLocalBase + q;
        if (ql >= N) continue;
        const int gq = base + ql;
        const int qb = q * NS;
        const int jl = idxLds[qb + s];
        const int j  = base + jl;
        const int c0 = cntLds[q];

        const float wv = ((s == 0) || (jl != idxLds[qb]))
                             ? (1.0f / (float)c0) : 0.0f;
        wOut[(size_t)gq * NS + s] = wv;

        const float* pj = pred + (size_t)j  * CCH;
        const float* pq = pred + (size_t)gq * CCH;
        const float* tj = tgt  + (size_t)j  * CCH;
        const float* tq = tgt  + (size_t)gq * CCH;
#pragma unroll
        for (int c = 0; c < CCH; ++c) {
            predOut[((size_t)gq * CCH + c) * NS + s] = pj[c] - pq[c];
            tgtOut [((size_t)gq * CCH + c) * NS + s] = tj[c] - tq[c];
        }
    }
}

extern "C" void kernel_launch(void* const* d_in, const int* in_sizes, int n_in,
                              void* d_out, int out_size, void* d_ws, size_t ws_size,
                              hipStream_t stream) {
    const float* xyz     = (const float*)d_in[0];
    const int*   xyz_cnt = (const int*)d_in[1]; (void)xyz_cnt;
    const float* pred    = (const float*)d_in[2];
    const float* tgt     = (const float*)d_in[3];
    const float* new_xyz = (const float*)d_in[4];

    const int nb   = in_sizes[1];              // B = 4
    const int ntot = in_sizes[0] / 3;          // 16384
    const int N    = ntot / nb;                // 4096 points per batch

    const int blocksPerBatch = (N + QPB - 1) / QPB;   // 32
    const int grid = nb * blocksPerBatch;             // 128

    ballquery_group_wmma<<<dim3(grid), dim3(256), 0, stream>>>(
        xyz, new_xyz, pred, tgt, (float*)d_out, N, nb, blocksPerBatch);
}